// InterfaceGraph_29678224016090
// MI455X (gfx1250) — compile-verified
//
#include <hip/hip_runtime.h>
#include <hip/hip_bf16.h>

typedef __attribute__((ext_vector_type(2))) float v2f;
typedef __attribute__((ext_vector_type(8))) float v8f;
typedef __attribute__((ext_vector_type(2))) int   v2i;

#define AS1 __attribute__((address_space(1)))
#define AS3 __attribute__((address_space(3)))

#define CUTOFF_F   10.0f
#define NUM_RES    2048
#define NGRAPH     64
#define PAD_COORD  1.0e12f
#define PAD_MAG    3.0e24f
#define BIG_MIN    1.0e30f

#define CHUNK      2048          // staged columns per graph (32 KB LDS)
#define BLOCK_THREADS 256        // 8 waves (wave32)
#define WAVES_PER_BLOCK (BLOCK_THREADS / 32)
#define SPLIT      2             // blocks per (side, graph)
#define WAVES_PER_GRAPH (WAVES_PER_BLOCK * SPLIT)

// Async global->LDS path (CDNA5): guarded so the file compiles on any toolchain.
#if defined(__has_builtin)
#if __has_builtin(__builtin_amdgcn_global_load_async_to_lds_b64) && \
    __has_builtin(__builtin_amdgcn_global_load_async_to_lds_b32)
#define HAVE_ASYNC_LDS 1
#endif
#endif

// ---------------------------------------------------------------------------
// ws layout (ints):
//  [0, NUM_RES)                    resMaskA
//  [NUM_RES, 2*NUM_RES)            resMaskB
//  [2*NUM_RES, +2*NGRAPH)          segA: {start,end} per graph
//  [2*NUM_RES+2*NGRAPH, +2*NGRAPH) segB
// ---------------------------------------------------------------------------

__device__ __forceinline__ int lower_bound_i32(const int* __restrict__ arr,
                                               int n, int key) {
  int lo = 0, hi = n;
  while (lo < hi) {
    int mid = (lo + hi) >> 1;
    if (arr[mid] < key) lo = mid + 1; else hi = mid;
  }
  return lo;
}

__global__ void init_kernel(int* __restrict__ ws,
                            const int* __restrict__ n2g_a, int Na,
                            const int* __restrict__ n2g_b, int Nb) {
  int tid = blockIdx.x * blockDim.x + threadIdx.x;
  int* resA = ws;
  int* resB = ws + NUM_RES;
  int* segA = ws + 2 * NUM_RES;
  int* segB = segA + 2 * NGRAPH;
  if (tid < NUM_RES) {
    resA[tid] = 0;
  } else if (tid < 2 * NUM_RES) {
    resB[tid - NUM_RES] = 0;
  } else if (tid < 2 * NUM_RES + NGRAPH) {
    int g = tid - 2 * NUM_RES;
    segA[2 * g]     = lower_bound_i32(n2g_a, Na, g);
    segA[2 * g + 1] = lower_bound_i32(n2g_a, Na, g + 1);
  } else if (tid < 2 * NUM_RES + 2 * NGRAPH) {
    int g = tid - (2 * NUM_RES + NGRAPH);
    segB[2 * g]     = lower_bound_i32(n2g_b, Nb, g);
    segB[2 * g + 1] = lower_bound_i32(n2g_b, Nb, g + 1);
  }
}

// ---------------------------------------------------------------------------
// Fused both-sides NN kernel. grid.x = 2 * NGRAPH * SPLIT blocks of 256.
// Stage target segment of graph g into LDS via async global->LDS copies,
// fix up w=|b|^2 + 16-column padding, then each wave sweeps 16-row query
// tiles with V_WMMA_F32_16X16X4_F32:
//   d2[M][N] = (-2 q_M) . t_N + |t_N|^2 + C   with C[r] = |q_M|^2
// B operand comes from a single ds_load_b64 per tile: lanes 0-15 read (x,y),
// lanes 16-31 read (z,|b|^2) via a half-wave base-address offset.
// Running min per row (v_min3_num_f32), shuffle-min, sqrt, atomicOr.
// ---------------------------------------------------------------------------
__global__ __launch_bounds__(BLOCK_THREADS)
void nn_kernel(const float* __restrict__ posA, const float* __restrict__ posB,
               const int* __restrict__ segA, const int* __restrict__ segB,
               const int* __restrict__ a2rA, const int* __restrict__ a2rB,
               int* __restrict__ resA, int* __restrict__ resB,
               int Na, int Nb, float* __restrict__ out) {
  __shared__ float ldsPos[CHUNK * 4];   // 16 B per column: x, y, z, |b|^2

  const int side = blockIdx.x / (NGRAPH * SPLIT);
  const int rem  = blockIdx.x % (NGRAPH * SPLIT);
  const int g    = rem / SPLIT;
  const int part = rem % SPLIT;

  const float* posQ = side ? posB : posA;
  const float* posT = side ? posA : posB;
  const int*   segQ = side ? segB : segA;
  const int*   segT = side ? segA : segB;
  const int*   a2rQ = side ? a2rB : a2rA;
  int*         resQ = side ? resB : resA;
  float*       distQ = out + (Na + Nb) + (side ? Na : 0);

  const int sQ = segQ[2 * g], eQ = segQ[2 * g + 1];
  const int sT = segT[2 * g], eT = segT[2 * g + 1];
  const int nSeg      = eT - sT;
  const int nStage    = nSeg < CHUNK ? nSeg : CHUNK;
  const int nStagePad = (nStage + 15) & ~15;   // <= CHUNK (CHUNK % 16 == 0)

  // ---- Stage target columns into LDS (async DMA, ASYNCcnt-tracked) ------
#if HAVE_ASYNC_LDS
  for (int c = threadIdx.x; c < nStage; c += BLOCK_THREADS) {
    const float* gp = posT + (size_t)(sT + c) * 3;
    float* lp = &ldsPos[c * 4];
    __builtin_amdgcn_global_load_async_to_lds_b64(
        (AS1 v2i*)gp, (AS3 v2i*)lp, 0, 0);                 // x, y
    __builtin_amdgcn_global_load_async_to_lds_b32(
        (AS1 int*)(gp + 2), (AS3 int*)(lp + 2), 0, 0);     // z
  }
#if __has_builtin(__builtin_amdgcn_s_wait_asynccnt)
  __builtin_amdgcn_s_wait_asynccnt(0);
#else
  asm volatile("s_wait_asynccnt 0" ::: "memory");
#endif
#else
  for (int c = threadIdx.x; c < nStage; c += BLOCK_THREADS) {
    const float* gp = posT + (size_t)(sT + c) * 3;
    ldsPos[c * 4 + 0] = gp[0];
    ldsPos[c * 4 + 1] = gp[1];
    ldsPos[c * 4 + 2] = gp[2];
  }
#endif
  __syncthreads();

  // ---- Fixup pass: w = |b|^2 for real columns, pad sentinels for tail ---
  for (int c = threadIdx.x; c < nStagePad; c += BLOCK_THREADS) {
    if (c < nStage) {
      const float x = ldsPos[c * 4 + 0];
      const float y = ldsPos[c * 4 + 1];
      const float z = ldsPos[c * 4 + 2];
      ldsPos[c * 4 + 3] = x * x + y * y + z * z;
    } else {
      float4 padv = make_float4(PAD_COORD, PAD_COORD, PAD_COORD, PAD_MAG);
      *(float4*)&ldsPos[c * 4] = padv;
    }
  }
  __syncthreads();

  // ---- Per-wave sweep over 16-row query tiles ---------------------------
  const int lane = threadIdx.x & 31;
  const int half = lane >> 4;
  const int l16  = lane & 15;
  const int wgWave = part * WAVES_PER_BLOCK + (threadIdx.x >> 5);

  const int nRows  = eQ - sQ;
  const int nTiles = (nRows + 15) >> 4;

  // Per-lane B pointer: lanes 0-15 -> (x,y) pair, lanes 16-31 -> (z,w) pair.
  const float* bbase = ldsPos + l16 * 4 + (half ? 2 : 0);

  for (int t = wgWave; t < nTiles; t += WAVES_PER_GRAPH) {
    const int row0 = sQ + t * 16;

    // A operand: row vector (-2x, -2y, -2z, 1)
    int rowL = row0 + l16;
    rowL = rowL < eQ ? rowL : eQ - 1;
    const float qx = posQ[rowL * 3 + 0];
    const float qy = posQ[rowL * 3 + 1];
    const float qz = posQ[rowL * 3 + 2];
    v2f A;
    A.x = half ? (-2.0f * qz) : (-2.0f * qx);
    A.y = half ? (1.0f)       : (-2.0f * qy);

    // C operand = |q_M|^2 per row: D = A*B + C yields d2 directly
    v8f amagv;
    float minv[8];
#pragma unroll
    for (int r = 0; r < 8; ++r) {
      int row = row0 + r + 8 * half;
      row = row < eQ ? row : eQ - 1;
      const float x = posQ[row * 3 + 0];
      const float y = posQ[row * 3 + 1];
      const float z = posQ[row * 3 + 2];
      amagv[r] = x * x + y * y + z * z;
      minv[r] = BIG_MIN;
    }

    // -- staged (LDS) columns: 2 tiles per iteration, both loads in flight
    int ct = 0;
    for (; ct + 32 <= nStagePad; ct += 32) {
      const v2f B0 = *(const v2f*)(bbase + ct * 4);          // ds_load_b64
      const v2f B1 = *(const v2f*)(bbase + (ct + 16) * 4);   // ds_load_b64

      const v8f acc0 = __builtin_amdgcn_wmma_f32_16x16x4_f32(
          false, A, false, B0, (short)0, amagv, false, false);
      const v8f acc1 = __builtin_amdgcn_wmma_f32_16x16x4_f32(
          false, A, false, B1, (short)0, amagv, false, false);

#pragma unroll
      for (int r = 0; r < 8; ++r) {
        minv[r] = fminf(minv[r], fminf(acc0[r], acc1[r]));
      }
    }
    for (; ct < nStagePad; ct += 16) {
      const v2f B = *(const v2f*)(bbase + ct * 4);           // ds_load_b64
      const v8f acc = __builtin_amdgcn_wmma_f32_16x16x4_f32(
          false, A, false, B, (short)0, amagv, false, false);
#pragma unroll
      for (int r = 0; r < 8; ++r) {
        minv[r] = fminf(minv[r], acc[r]);
      }
    }

    // -- rare oversize-segment remainder straight from global memory
    for (int cb = sT + CHUNK; cb < eT; cb += 16) {
      const int col  = cb + l16;
      const int colc = col < eT ? col : eT - 1;
      float bx = posT[colc * 3 + 0];
      float by = posT[colc * 3 + 1];
      float bz = posT[colc * 3 + 2];
      const bool pad = col >= eT;
      bx = pad ? PAD_COORD : bx;
      by = pad ? PAD_COORD : by;
      bz = pad ? PAD_COORD : bz;
      const float bmag = bx * bx + by * by + bz * bz;

      v2f B;
      B.x = half ? bz   : bx;
      B.y = half ? bmag : by;

      const v8f acc = __builtin_amdgcn_wmma_f32_16x16x4_f32(
          false, A, false, B, (short)0, amagv, false, false);
#pragma unroll
      for (int r = 0; r < 8; ++r) {
        minv[r] = fminf(minv[r], acc[r]);
      }
    }

    // -- min-reduce across the 16 lanes of each half-wave
#pragma unroll
    for (int r = 0; r < 8; ++r) {
      float v = minv[r];
      v = fminf(v, __shfl_xor(v, 1, 32));
      v = fminf(v, __shfl_xor(v, 2, 32));
      v = fminf(v, __shfl_xor(v, 4, 32));
      v = fminf(v, __shfl_xor(v, 8, 32));
      minv[r] = v;
    }

    if (l16 == 0) {
#pragma unroll
      for (int r = 0; r < 8; ++r) {
        const int row = row0 + r + 8 * half;
        if (row < eQ) {
          const float d = sqrtf(fmaxf(minv[r], 0.0f));
          distQ[row] = d;
          if (d < CUTOFF_F) {
            atomicOr(&resQ[a2rQ[row]], 1);
          }
        }
      }
    }
  }
}

// ---------------------------------------------------------------------------
// mask[i] = resFlag[residue[i]] | is_mutation[i]  (as 0.0f / 1.0f)
// ---------------------------------------------------------------------------
__global__ void mask_kernel(const int* __restrict__ resA,
                            const int* __restrict__ resB,
                            const int* __restrict__ a2rA,
                            const int* __restrict__ a2rB,
                            const unsigned char* __restrict__ mut,
                            int Na, int Nb,
                            float* __restrict__ maskOut) {
  const int i = blockIdx.x * blockDim.x + threadIdx.x;
  const int N = Na + Nb;
  if (i >= N) return;
  int m;
  if (i < Na) m = resA[a2rA[i]];
  else        m = resB[a2rB[i - Na]];
  m |= (int)mut[i];
  maskOut[i] = m ? 1.0f : 0.0f;
}

// ---------------------------------------------------------------------------
extern "C" void kernel_launch(void* const* d_in, const int* in_sizes, int n_in,
                              void* d_out, int out_size, void* d_ws, size_t ws_size,
                              hipStream_t stream) {
  const float* pos_a = (const float*)d_in[0];
  const float* pos_b = (const float*)d_in[1];
  const int*   n2g_a = (const int*)d_in[2];
  const int*   n2g_b = (const int*)d_in[3];
  const int*   a2r_a = (const int*)d_in[4];
  const int*   a2r_b = (const int*)d_in[5];
  const unsigned char* mut = (const unsigned char*)d_in[6];

  const int Na = in_sizes[2];
  const int Nb = in_sizes[3];

  int* ws   = (int*)d_ws;
  int* resA = ws;
  int* resB = ws + NUM_RES;
  int* segA = ws + 2 * NUM_RES;
  int* segB = segA + 2 * NGRAPH;

  float* out = (float*)d_out;   // [0,Na+Nb): mask ; [Na+Nb, 2(Na+Nb)): dists

  // 1) zero residue flags + binary-search graph segment boundaries
  const int initThreads = 2 * NUM_RES + 2 * NGRAPH;
  init_kernel<<<(initThreads + 255) / 256, 256, 0, stream>>>(ws, n2g_a, Na, n2g_b, Nb);

  // 2) fused both-sides segmented WMMA nearest-neighbor sweep
  nn_kernel<<<2 * NGRAPH * SPLIT, BLOCK_THREADS, 0, stream>>>(
      pos_a, pos_b, segA, segB, a2r_a, a2r_b, resA, resB, Na, Nb, out);

  // 3) broadcast residue interface flags back to atoms, OR with mutation bit
  mask_kernel<<<(Na + Nb + 255) / 256, 256, 0, stream>>>(resA, resB, a2r_a, a2r_b,
                                                         mut, Na, Nb, out);
}